// DeepFace_30142080483814
// MI455X (gfx1250) — compile-verified
//
#include <hip/hip_runtime.h>
#include <hip/hip_bf16.h>

typedef __attribute__((ext_vector_type(2))) float v2f;
typedef __attribute__((ext_vector_type(8))) float v8f;

#define WPB 8           // waves per 256-thread block
#define MAX_KPAD 2592   // conv2 K = 32*9*9

__device__ __forceinline__ int imin(int a, int b) { return a < b ? a : b; }

// Unified kernel: per output position, D[32x16] = A[32xK] x W[Kx16] via
// V_WMMA_F32_16X16X4_F32 (two 16-row batch tiles), bias + ReLU.
// lc_weights=0: shared conv weights (CoutT,K); lc_weights=1: per-position (OH,OW,CoutT,K).
// WVEC: weight rows 8B-aligned (K even) -> b64 weight loads.
template <bool WVEC>
__global__ __launch_bounds__(256) void lc_gemm_wmma(
    const float* __restrict__ in,   // (32, Cin, Hin, Win)
    const float* __restrict__ wgt,
    const float* __restrict__ bias, // bpp: (CoutT,OH,OW) else (CoutT)
    float* __restrict__ out,        // (32, CoutT, OH, OW)
    int Hin, int Win, int OH, int OW, int stride, int ksz,
    int Cin, int CoutT, int K, int lc_weights, int bias_per_pos)
{
    __shared__ int s_off[MAX_KPAD];   // BYTE offsets into one batch image; pad -> 0 (safe)

    const int kk2  = ksz * ksz;
    const int Kpad = (K + 3) & ~3;
    for (int kk = threadIdx.x; kk < Kpad; kk += blockDim.x) {
        int o = 0;
        if (kk < K) {
            int ci = kk / kk2, rem = kk % kk2;
            int kh = rem / ksz, kw = rem % ksz;
            o = ((ci * Hin + kh) * Win + kw) * 4;
        }
        s_off[kk] = o;
    }
    __syncthreads();

    const int npos = OH * OW;
    const int wave = threadIdx.x >> 5;
    const int lane = threadIdx.x & 31;
    const int pos  = blockIdx.x * WPB + wave;
    if (pos >= npos) return;                  // wave-uniform exit

    const int y = pos / OW, x = pos % OW;
    const int co_base = blockIdx.y * 16;
    const int lo16 = lane & 15;               // A row (batch) / B col (cout)
    const int h    = lane >> 4;               // K-half selector

    // A: uniform base pointer + per-lane 32-bit byte offsets
    const int inBatchB = Cin * Hin * Win * 4;
    const char* inw  = (const char*)in + (long)((y * stride) * Win + x * stride) * 4;
    const int  aoff0 = lo16 * inBatchB;
    const int  aoff1 = aoff0 + 16 * inBatchB;

    // B: uniform base + per-lane running byte offset (step 16B per K-step)
    const char* wbase = (const char*)wgt + (long)(lc_weights ? pos : 0) * CoutT * K * 4;
    int woff = ((co_base + lo16) * K + 2 * h) * 4;
    const int wend = ((co_base + lo16) * K + (K - 1)) * 4;  // last valid element (tail clamp)

    v8f acc0 = {}; v8f acc1 = {};
    const int Kmain = K & ~3;

    #pragma unroll 2
    for (int kk0 = 0; kk0 < Kmain; kk0 += 4) {
        const int2 o = *(const int2*)(&s_off[kk0 + 2 * h]);
        v2f a0, a1, bb;
        a0.x = *(const float*)(inw + (aoff0 + o.x));
        a0.y = *(const float*)(inw + (aoff0 + o.y));
        a1.x = *(const float*)(inw + (aoff1 + o.x));
        a1.y = *(const float*)(inw + (aoff1 + o.y));
        if (WVEC) {
            bb = *(const v2f*)(wbase + woff);
        } else {
            bb.x = *(const float*)(wbase + woff);
            bb.y = *(const float*)(wbase + woff + 4);
        }
        woff += 16;
        acc0 = __builtin_amdgcn_wmma_f32_16x16x4_f32(false, a0, false, bb,
                                                     (short)0, acc0, false, false);
        acc1 = __builtin_amdgcn_wmma_f32_16x16x4_f32(false, a1, false, bb,
                                                     (short)0, acc1, false, false);
    }

    if (Kmain < K) {   // tail K-step (conv1 only). Branch-free: clamp addr, mask value.
        const int k0 = Kmain + 2 * h;
        const int2 o = *(const int2*)(&s_off[k0]);   // padded entries are 0 (valid)
        v2f a0, a1, bb;
        a0.x = *(const float*)(inw + (aoff0 + o.x));
        a0.y = *(const float*)(inw + (aoff0 + o.y));
        a1.x = *(const float*)(inw + (aoff1 + o.x));
        a1.y = *(const float*)(inw + (aoff1 + o.y));
        float w0 = *(const float*)(wbase + imin(woff,     wend));
        float w1 = *(const float*)(wbase + imin(woff + 4, wend));
        bb.x = (k0     < K) ? w0 : 0.0f;
        bb.y = (k0 + 1 < K) ? w1 : 0.0f;
        acc0 = __builtin_amdgcn_wmma_f32_16x16x4_f32(false, a0, false, bb,
                                                     (short)0, acc0, false, false);
        acc1 = __builtin_amdgcn_wmma_f32_16x16x4_f32(false, a1, false, bb,
                                                     (short)0, acc1, false, false);
    }

    const int co = co_base + lo16;
    const float bv = bias_per_pos ? bias[(long)co * npos + pos] : bias[co];
    #pragma unroll
    for (int r = 0; r < 8; ++r) {
        const int b0 = r + 8 * h;             // C/D layout: M = r + 8*(lane/16)
        float v0 = acc0[r] + bv; v0 = fmaxf(v0, 0.0f);
        out[(((long)b0 * CoutT + co) * OH + y) * OW + x] = v0;
        const int b1 = b0 + 16;
        float v1 = acc1[r] + bv; v1 = fmaxf(v1, 0.0f);
        out[(((long)b1 * CoutT + co) * OH + y) * OW + x] = v1;
    }
}

// 3x3 stride-2 max pool, ceil mode (142 -> 71), -inf pad bottom/right.
__global__ __launch_bounds__(256) void maxpool3s2(
    const float* __restrict__ in, float* __restrict__ out,
    int Hin, int Win, int OH, int OW, int total)
{
    int idx = blockIdx.x * blockDim.x + threadIdx.x;
    if (idx >= total) return;
    const int ow = idx % OW;
    const int oh = (idx / OW) % OH;
    const int nc = idx / (OW * OH);
    const float* p = in + (long)nc * Hin * Win;
    float m = -3.402823466e38f;
    #pragma unroll
    for (int r = 0; r < 3; ++r) {
        int ih = oh * 2 + r;
        if (ih >= Hin) continue;
        #pragma unroll
        for (int c = 0; c < 3; ++c) {
            int iw = ow * 2 + c;
            if (iw >= Win) continue;
            m = fmaxf(m, p[ih * Win + iw]);
        }
    }
    out[idx] = m;
}

// FC: out[32,N] = A[32,Kt] x W[N,Kt]^T + bias. One wave per 16-column tile.
__global__ __launch_bounds__(256) void fc_wmma(
    const float* __restrict__ A, const float* __restrict__ W,
    const float* __restrict__ bias, float* __restrict__ out, int Kt, int N)
{
    const int wave = threadIdx.x >> 5;
    const int lane = threadIdx.x & 31;
    const int ct   = blockIdx.x * WPB + wave;
    if (ct * 16 >= N) return;

    const int lo16 = lane & 15;
    const int h    = lane >> 4;
    const char* Ab = (const char*)A;
    const char* Wb = (const char*)W;
    int ao0 = lo16 * Kt * 4 + 8 * h;             // per-lane byte offsets, step 16B
    int ao1 = ao0 + 16 * Kt * 4;
    int wo  = (ct * 16 + lo16) * Kt * 4 + 8 * h; // < 116MB, fits int32

    v8f acc0 = {}; v8f acc1 = {};
    #pragma unroll 2
    for (int it = 0; it < Kt; it += 4) {         // Kt = 7056, multiple of 4
        v2f a0 = *(const v2f*)(Ab + ao0);
        v2f a1 = *(const v2f*)(Ab + ao1);
        v2f bb = *(const v2f*)(Wb + wo);
        ao0 += 16; ao1 += 16; wo += 16;
        acc0 = __builtin_amdgcn_wmma_f32_16x16x4_f32(false, a0, false, bb,
                                                     (short)0, acc0, false, false);
        acc1 = __builtin_amdgcn_wmma_f32_16x16x4_f32(false, a1, false, bb,
                                                     (short)0, acc1, false, false);
    }

    const int col = ct * 16 + lo16;
    const float bv = bias[col];
    #pragma unroll
    for (int r = 0; r < 8; ++r) {
        const int b0 = r + 8 * h;
        out[(long)b0 * N + col]        = acc0[r] + bv;
        out[(long)(b0 + 16) * N + col] = acc1[r] + bv;
    }
}

extern "C" void kernel_launch(void* const* d_in, const int* in_sizes, int n_in,
                              void* d_out, int out_size, void* d_ws, size_t ws_size,
                              hipStream_t stream) {
    const float* x   = (const float*)d_in[0];
    const float* c1w = (const float*)d_in[1];
    const float* c1b = (const float*)d_in[2];
    const float* c2w = (const float*)d_in[3];
    const float* c2b = (const float*)d_in[4];
    const float* l1w = (const float*)d_in[5];
    const float* l1b = (const float*)d_in[6];
    const float* l2w = (const float*)d_in[7];
    const float* l2b = (const float*)d_in[8];
    const float* l3w = (const float*)d_in[9];
    const float* l3b = (const float*)d_in[10];
    const float* fcw = (const float*)d_in[11];
    const float* fcb = (const float*)d_in[12];
    float* out = (float*)d_out;

    float* ws = (float*)d_ws;
    float* h1 = ws;                    // (32,32,142,142) = 20,647,936
    float* p1 = h1 + 20647936;         // (32,32,71,71)   =  5,161,984
    float* h2 = p1 + 5161984;          // (32,16,63,63)   =  2,032,128
    float* h3 = h2 + 2032128;          // (32,16,55,55)   =  1,548,800
    float* h4 = h3 + 1548800;          // (32,16,25,25)   =    320,000
    float* h5 = h4 + 320000;           // (32,16,21,21)   =    225,792

    // conv1 + relu: (32,1,152,152) -> (32,32,142,142); Cout=32 -> grid.y=2; K=121 odd -> scalar weights
    {
        int npos = 142 * 142;
        dim3 g((npos + WPB - 1) / WPB, 2);
        lc_gemm_wmma<false><<<g, 256, 0, stream>>>(x, c1w, c1b, h1,
            152, 152, 142, 142, 1, 11, 1, 32, 121, 0, 0);
    }
    // maxpool 3x3 s2 ceil: -> (32,32,71,71)
    {
        int total = 32 * 32 * 71 * 71;
        maxpool3s2<<<(total + 255) / 256, 256, 0, stream>>>(h1, p1, 142, 142, 71, 71, total);
    }
    // conv2 + relu: -> (32,16,63,63)
    {
        int npos = 63 * 63;
        dim3 g((npos + WPB - 1) / WPB, 1);
        lc_gemm_wmma<true><<<g, 256, 0, stream>>>(p1, c2w, c2b, h2,
            71, 71, 63, 63, 1, 9, 32, 16, 2592, 0, 0);
    }
    // lc1 + relu: -> (32,16,55,55)
    {
        int npos = 55 * 55;
        dim3 g((npos + WPB - 1) / WPB, 1);
        lc_gemm_wmma<true><<<g, 256, 0, stream>>>(h2, l1w, l1b, h3,
            63, 63, 55, 55, 1, 9, 16, 16, 1296, 1, 1);
    }
    // lc2 + relu (stride 2): -> (32,16,25,25)
    {
        int npos = 25 * 25;
        dim3 g((npos + WPB - 1) / WPB, 1);
        lc_gemm_wmma<true><<<g, 256, 0, stream>>>(h3, l2w, l2b, h4,
            55, 55, 25, 25, 2, 7, 16, 16, 784, 1, 1);
    }
    // lc3 + relu: -> (32,16,21,21)
    {
        int npos = 21 * 21;
        dim3 g((npos + WPB - 1) / WPB, 1);
        lc_gemm_wmma<true><<<g, 256, 0, stream>>>(h4, l3w, l3b, h5,
            25, 25, 21, 21, 1, 5, 16, 16, 400, 1, 1);
    }
    // fc: (32,7056) @ (4096,7056)^T + b
    fc_wmma<<<(4096 / 16 + WPB - 1) / WPB, 256, 0, stream>>>(h5, fcw, fcb, out, 7056, 4096);
}